// deformable_ConvBNPReLU_63685775065197
// MI455X (gfx1250) — compile-verified
//
#include <hip/hip_runtime.h>

// ---------------------------------------------------------------------------
// Deformable Conv + BN + PReLU for MI455X (gfx1250, wave32, WMMA + TDM).
// fp32 end-to-end; GEMMs on V_WMMA_F32_16X16X4_F32; B-tiles DMA'd into LDS
// by the Tensor Data Mover (tensor_load_to_lds + s_wait_tensorcnt).
//
// Shapes: B=8, Cin=Cout=64, H=W=128, K=3, kk=9.
//   offset conv : M=131072, N=27(pad 32), K=576
//   deform conv : M=131072, N=64,          K=576 (A gathered bilinearly)
//
// Workspace layout (floats):
//   om    : 8*27*16384 = 3,538,944
//   preBN : 8*64*16384 = 8,388,608
//   stats : 128  (sum[64], sumsq[64])
//   wot   : repacked offset weights [k][cin][32pad] = 18,432
//   wt    : repacked deform weights [k][cin][cout]  = 36,864
// ---------------------------------------------------------------------------

typedef float v2f __attribute__((ext_vector_type(2)));
typedef float v8f __attribute__((ext_vector_type(8)));

#define BB   8
#define CIN  64
#define COUT 64
#define HH   128
#define WW   128
#define HW   (HH * WW)          // 16384
#define KK   9
#define NPIX (BB * HW)          // 131072
#define OM_CH 27
#define OM_ELEMS  ((size_t)BB * OM_CH * HW)
#define OUT_ELEMS ((size_t)BB * COUT * HW)
#define WOT_ELEMS (KK * CIN * 32)    // 18432
#define WT_ELEMS  (KK * CIN * COUT)  // 36864

// ------------------------------------------------ Tensor Data Mover helper
#if __has_builtin(__builtin_amdgcn_tensor_load_to_lds) && \
    __has_builtin(__builtin_amdgcn_s_wait_tensorcnt)
#define HAVE_TDM 1
typedef unsigned int u32x4 __attribute__((ext_vector_type(4)));
typedef int          i32x8 __attribute__((ext_vector_type(8)));
typedef int          i32x4 __attribute__((ext_vector_type(4)));

// 1-D contiguous DMA of `ndw` dwords global -> LDS (ndw < 65536).
// D# per CDNA5 ISA §8: group0 = {count, lds_addr, global_addr, type=2},
// group1 = {data_size=4B, tensor_dim0=tile_dim0=ndw, rest 0}.
// This toolchain exposes the 6-arg builtin:
//   (u32x4 g0, i32x8 g1, i32x4 g2, i32x4 g3, i32x8 extra, i32 cpol)
__device__ __forceinline__ void tdm_load_1d(const float* gsrc, void* lds_dst,
                                            unsigned ndw) {
    const unsigned long long ga = (unsigned long long)(uintptr_t)gsrc;
    // generic pointer to __shared__: low 32 bits are the LDS byte offset
    const unsigned lds_off = (unsigned)(uintptr_t)lds_dst;
    u32x4 g0;
    g0[0] = 1u;                                        // count=1, user desc
    g0[1] = lds_off;                                   // lds_addr (bytes)
    g0[2] = (unsigned)(ga & 0xffffffffu);              // global_addr[31:0]
    g0[3] = (unsigned)((ga >> 32) & 0x01ffffffu)       // global_addr[56:32]
          | (2u << 30);                                // type = 2 ("image")
    i32x8 g1;
    g1[0] = (int)(2u << 16);                           // data_size=2 -> 4 bytes
    g1[1] = (int)((ndw & 0xffffu) << 16);              // tensor_dim0[15:0]
    g1[2] = (int)((ndw >> 16) & 0xffffu);              // tensor_dim0[31:16]
    g1[3] = (int)((ndw & 0xffffu) << 16);              // tile_dim0
    g1[4] = 0; g1[5] = 0; g1[6] = 0; g1[7] = 0;        // tile_dim1/2=0, strides
    const i32x4 z4 = {0, 0, 0, 0};
    const i32x8 z8 = {0, 0, 0, 0, 0, 0, 0, 0};
    __builtin_amdgcn_tensor_load_to_lds(g0, g1, z4, z4, z8, 0);
}
#endif

// ---------------------------------------------------------------- stats zero
__global__ void zero_stats_kernel(float* stats) {
    if (threadIdx.x < 128) stats[threadIdx.x] = 0.0f;
}

// ------------------------------------------------ weight repack (tiny)
// wt [k][cin][cout]   <- wdef (cout, cin, 3, 3)
// wot[k][cin][och32]  <- wofs (27, cin, 3, 3), zero-padded och 27..31
__global__ void repack_weights(const float* __restrict__ wofs,
                               const float* __restrict__ wdef,
                               float* __restrict__ wot,
                               float* __restrict__ wt) {
    const int i = blockIdx.x * 256 + threadIdx.x;
    if (i < WT_ELEMS) {
        const int k = i >> 12, cin = (i >> 6) & 63, co = i & 63;
        wt[i] = wdef[((size_t)co * CIN + cin) * KK + k];
    }
    if (i < WOT_ELEMS) {
        const int k = i >> 11, cin = (i >> 5) & 63, och = i & 31;
        wot[i] = (och < OM_CH) ? wofs[((size_t)och * CIN + cin) * KK + k] : 0.0f;
    }
}

// ------------------------------------------------------- offset conv (WMMA)
// Grid: NPIX/64 blocks, 256 threads (8 waves). Wave tile: 16 pix x 16 och.
__global__ void offset_conv_wmma(const float* __restrict__ x,
                                 const float* __restrict__ wot,
                                 const float* __restrict__ bofs,
                                 float* __restrict__ om) {
    __shared__ float As[64][68];   // [pixel][cin], padded stride
    __shared__ float Bs[64][32];   // [cin][och]  (contiguous = repacked tap)

    const int tid  = threadIdx.x;
    const int wave = tid >> 5;
    const int lane = tid & 31;
    const int p0   = blockIdx.x * 64;

    // staging role: pixel = tid/4, 16-channel chunk = tid%4
    const int spix = tid >> 2;
    const int sc0  = (tid & 3) * 16;
    const int sp   = p0 + spix;
    const int sb   = sp >> 14;
    const int srem = sp & 16383;
    const int sy   = srem >> 7;
    const int sx   = srem & 127;

    const int mt   = wave >> 1;      // pixel tile 0..3
    const int nt   = wave & 1;       // och tile 0/1
    const int row  = lane & 15;
    const int half = lane >> 4;

    v8f acc = {};

    for (int k = 0; k < KK; ++k) {
        const int ky = k / 3 - 1, kx = k % 3 - 1;
#ifdef HAVE_TDM
        if (wave == 0) tdm_load_1d(wot + (size_t)k * CIN * 32, &Bs[0][0], CIN * 32);
#endif
        // --- stage A: shifted patch, zero-padded borders
        {
            const int yy = sy + ky, xx = sx + kx;
            const bool ok = (yy >= 0) & (yy < HH) & (xx >= 0) & (xx < WW);
            const float* src = x + (size_t)sb * CIN * HW + (ok ? (yy * WW + xx) : 0);
#pragma unroll
            for (int j = 0; j < 16; ++j) {
                const int c = sc0 + j;
                As[spix][c] = ok ? src[(size_t)c * HW] : 0.0f;
            }
        }
#ifndef HAVE_TDM
#pragma unroll
        for (int j = 0; j < 8; ++j) {
            const int idx = tid * 8 + j;   // 2048 contiguous entries
            (&Bs[0][0])[idx] = wot[(size_t)k * CIN * 32 + idx];
        }
#endif
#ifdef HAVE_TDM
        if (wave == 0) __builtin_amdgcn_s_wait_tensorcnt(0);
#endif
        __syncthreads();
        // --- 16 K-steps of f32 WMMA (K=4 each)
#pragma unroll
        for (int c0 = 0; c0 < CIN; c0 += 4) {
            const int cA = c0 + half * 2;
            v2f a; a.x = As[mt * 16 + row][cA];      a.y = As[mt * 16 + row][cA + 1];
            v2f b; b.x = Bs[cA][nt * 16 + row];      b.y = Bs[cA + 1][nt * 16 + row];
            acc = __builtin_amdgcn_wmma_f32_16x16x4_f32(
                false, a, false, b, (short)0, acc, false, false);
        }
        __syncthreads();
    }

    // epilogue: bias + store om[b][och][y][x]
    const int och = nt * 16 + row;
    if (och < OM_CH) {
        const float bias = bofs[och];
#pragma unroll
        for (int r = 0; r < 8; ++r) {
            const int m = mt * 16 + r + half * 8;   // C layout: VGPR r -> M=r / r+8
            const int p = p0 + m;
            const int b = p >> 14, rem = p & 16383;
            om[((size_t)b * OM_CH + och) * HW + rem] = acc[r] + bias;
        }
    }
}

// ------------------------------------------------ deformable conv (WMMA)
// Grid: NPIX/64 blocks, 256 threads (8 waves). Wave tile: 32 pix x 16 cout
// (two 16x16 accumulators sharing each B fragment).
__global__ void deform_conv_wmma(const float* __restrict__ x,
                                 const float* __restrict__ om,
                                 const float* __restrict__ wt,
                                 float* __restrict__ out) {
    __shared__ float As[64][68];   // [pixel][cin] bilinearly sampled
    __shared__ float Bs[64][64];   // [cin][cout] (contiguous = repacked tap)

    const int tid  = threadIdx.x;
    const int wave = tid >> 5;
    const int lane = tid & 31;
    const int p0   = blockIdx.x * 64;

    // staging role: pixel = tid/4, 16-channel chunk = tid%4
    const int spix = tid >> 2;
    const int sc0  = (tid & 3) * 16;
    const int sp   = p0 + spix;
    const int sb   = sp >> 14;
    const int srem = sp & 16383;
    const int sy   = srem >> 7;
    const int sx   = srem & 127;

    const int nt   = wave & 3;    // cout tile 0..3
    const int mg   = wave >> 2;   // pixel-tile pair 0/1 (pixels mg*32 .. +31)
    const int row  = lane & 15;
    const int half = lane >> 4;

    const float* omb = om + (size_t)sb * OM_CH * HW + srem;
    const float* xb  = x  + (size_t)sb * CIN * HW;

    v8f acc0 = {};
    v8f acc1 = {};

    for (int k = 0; k < KK; ++k) {
#ifdef HAVE_TDM
        if (wave == 0) tdm_load_1d(wt + (size_t)k * CIN * COUT, &Bs[0][0], CIN * COUT);
#endif
        // --- per-pixel deformable bilinear sample of 16 channels into LDS
        {
            const float oy = omb[(size_t)(2 * k) * HW];
            const float ox = omb[(size_t)(2 * k + 1) * HW];
            const float mk = omb[(size_t)(18 + k) * HW];
            const float ms = 1.0f / (1.0f + __expf(-mk));     // sigmoid(mask)

            const float py = (float)(sy - 1 + k / 3) + oy;
            const float px = (float)(sx - 1 + k % 3) + ox;
            const float y0f = floorf(py), x0f = floorf(px);
            const float wy = py - y0f, wx = px - x0f;
            const int y0 = (int)y0f, x0 = (int)x0f;
            const int y1 = y0 + 1,   x1 = x0 + 1;

            const bool vy0 = (y0 >= 0) & (y0 < HH), vy1 = (y1 >= 0) & (y1 < HH);
            const bool vx0 = (x0 >= 0) & (x0 < WW), vx1 = (x1 >= 0) & (x1 < WW);
            const float w00 = (1.f - wy) * (1.f - wx) * ms * (float)(vy0 & vx0);
            const float w01 = (1.f - wy) * wx         * ms * (float)(vy0 & vx1);
            const float w10 = wy * (1.f - wx)         * ms * (float)(vy1 & vx0);
            const float w11 = wy * wx                 * ms * (float)(vy1 & vx1);

            const int yc0 = min(max(y0, 0), HH - 1), yc1 = min(max(y1, 0), HH - 1);
            const int xc0 = min(max(x0, 0), WW - 1), xc1 = min(max(x1, 0), WW - 1);
            const int i00 = yc0 * WW + xc0, i01 = yc0 * WW + xc1;
            const int i10 = yc1 * WW + xc0, i11 = yc1 * WW + xc1;
#pragma unroll
            for (int j = 0; j < 16; ++j) {
                const int c = sc0 + j;
                const float* xc = xb + (size_t)c * HW;
                As[spix][c] = w00 * xc[i00] + w01 * xc[i01]
                            + w10 * xc[i10] + w11 * xc[i11];
            }
        }
#ifndef HAVE_TDM
#pragma unroll
        for (int j = 0; j < 16; ++j) {
            const int idx = tid * 16 + j;  // 4096 contiguous entries
            (&Bs[0][0])[idx] = wt[(size_t)k * CIN * COUT + idx];
        }
#endif
#ifdef HAVE_TDM
        if (wave == 0) __builtin_amdgcn_s_wait_tensorcnt(0);
#endif
        __syncthreads();
        // --- 16 K-steps of f32 WMMA; B fragment shared by both M-tiles
#pragma unroll
        for (int c0 = 0; c0 < CIN; c0 += 4) {
            const int cA = c0 + half * 2;
            v2f b;  b.x  = Bs[cA][nt * 16 + row];        b.y  = Bs[cA + 1][nt * 16 + row];
            v2f a0; a0.x = As[mg * 32 + row][cA];        a0.y = As[mg * 32 + row][cA + 1];
            v2f a1; a1.x = As[mg * 32 + 16 + row][cA];   a1.y = As[mg * 32 + 16 + row][cA + 1];
            acc0 = __builtin_amdgcn_wmma_f32_16x16x4_f32(
                false, a0, false, b, (short)0, acc0, false, false);
            acc1 = __builtin_amdgcn_wmma_f32_16x16x4_f32(
                false, a1, false, b, (short)0, acc1, false, false);
        }
        __syncthreads();
    }

    // epilogue: write pre-BN output (NCHW)
    const int co = nt * 16 + row;
#pragma unroll
    for (int r = 0; r < 8; ++r) {
        const int m0 = mg * 32 + r + half * 8;
        const int m1 = m0 + 16;
        const int pA = p0 + m0, pB = p0 + m1;
        out[((size_t)(pA >> 14) * COUT + co) * HW + (pA & 16383)] = acc0[r];
        out[((size_t)(pB >> 14) * COUT + co) * HW + (pB & 16383)] = acc1[r];
    }
}

// -------------------------------------------------- per-channel sum / sumsq
// Grid: 64 channels * 32 slabs = 2048 blocks, 256 threads, 16 elems/thread.
__global__ void channel_stats(const float* __restrict__ out,
                              float* __restrict__ stats) {
    __shared__ float s1[256];
    __shared__ float s2[256];
    const int c    = blockIdx.x >> 5;
    const int slab = blockIdx.x & 31;
    const int tid  = threadIdx.x;

    float a = 0.f, b = 0.f;
    const int base = slab * 4096 + tid * 16;
#pragma unroll
    for (int j = 0; j < 16; ++j) {
        const int e  = base + j;            // 0..131071 within channel
        const int bb = e >> 14, hw = e & 16383;
        const float v = out[((size_t)bb * COUT + c) * HW + hw];
        a += v; b += v * v;
    }
    s1[tid] = a; s2[tid] = b;
    __syncthreads();
    for (int s = 128; s > 0; s >>= 1) {
        if (tid < s) { s1[tid] += s1[tid + s]; s2[tid] += s2[tid + s]; }
        __syncthreads();
    }
    if (tid == 0) {
        atomicAdd(&stats[c],      s1[0]);
        atomicAdd(&stats[64 + c], s2[0]);
    }
}

// -------------------------------------------------------- BN + PReLU
__global__ void bn_prelu(const float* __restrict__ out,
                         const float* __restrict__ stats,
                         const float* __restrict__ gamma,
                         const float* __restrict__ beta,
                         const float* __restrict__ alpha,
                         float* __restrict__ y) {
    const size_t idx = (size_t)blockIdx.x * blockDim.x + threadIdx.x;
    if (idx >= OUT_ELEMS) return;
    const int c = (int)((idx >> 14) & 63);
    const float inv_n = 1.0f / (float)NPIX;
    const float mean  = stats[c] * inv_n;
    const float var   = stats[64 + c] * inv_n - mean * mean;
    const float sc    = gamma[c] * rsqrtf(var + 1e-3f);
    const float v     = (out[idx] - mean) * sc + beta[c];
    y[idx] = v > 0.0f ? v : alpha[c] * v;
}

// ---------------------------------------------------------------------------
extern "C" void kernel_launch(void* const* d_in, const int* in_sizes, int n_in,
                              void* d_out, int out_size, void* d_ws, size_t ws_size,
                              hipStream_t stream) {
    const float* x      = (const float*)d_in[0];   // (8,64,128,128)
    const float* wofs   = (const float*)d_in[1];   // (27,64,3,3)
    const float* bofs   = (const float*)d_in[2];   // (27,)
    const float* wdef   = (const float*)d_in[3];   // (64,64,3,3)
    const float* bgamma = (const float*)d_in[4];   // (64,)
    const float* bbeta  = (const float*)d_in[5];   // (64,)
    const float* palpha = (const float*)d_in[6];   // (64,)
    float*       yout   = (float*)d_out;

    float* ws    = (float*)d_ws;
    float* om    = ws;
    float* preBN = ws + OM_ELEMS;
    float* stats = ws + OM_ELEMS + OUT_ELEMS;
    float* wot   = stats + 128;
    float* wt    = wot + WOT_ELEMS;

    zero_stats_kernel<<<1, 128, 0, stream>>>(stats);

    repack_weights<<<(WT_ELEMS + 255) / 256, 256, 0, stream>>>(wofs, wdef, wot, wt);

    offset_conv_wmma<<<NPIX / 64, 256, 0, stream>>>(x, wot, bofs, om);

    deform_conv_wmma<<<NPIX / 64, 256, 0, stream>>>(x, om, wt, preBN);

    channel_stats<<<COUT * 32, 256, 0, stream>>>(preBN, stats);

    const int n = (int)OUT_ELEMS;
    bn_prelu<<<(n + 255) / 256, 256, 0, stream>>>(preBN, stats, bgamma, bbeta,
                                                  palpha, yout);
}